// CombineModel_wo_net_20031727468607
// MI455X (gfx1250) — compile-verified
//
#include <hip/hip_runtime.h>

// ---------------------------------------------------------------------------
// Problem constants (from the reference)
// ---------------------------------------------------------------------------
#define BATCH   16
#define CH      3
#define HW      (544 * 960)      // 522240 pixels per channel
#define HW4     (HW / 4)         // 130560 float4 per channel
#define BPB     128              // blocks per batch (K dim of the WMMA reduce)
#define NPART   (BATCH * BPB)    // 2048 partials per plane
#define TPB     256              // 8 waves of 32

typedef float v2f __attribute__((ext_vector_type(2)));
typedef float v8f __attribute__((ext_vector_type(8)));

// ---------------------------------------------------------------------------
// Kernel 1: bandwidth-bound streaming reduce.
// Each block owns (batch, blk) and strides over the 130560 float4 of its
// batch, loading all 3 channels with coalesced global_load_b128.
// Emits 3 per-block partials (gray-sum, bright-count, dark-count) into ws
// laid out as three 16x128 row-major matrices: plane[batch*BPB + blk].
// ---------------------------------------------------------------------------
__global__ __launch_bounds__(TPB) void reduce_kernel(
    const float* __restrict__ img, float* __restrict__ ws) {
  const int batch = blockIdx.x / BPB;
  const int blk   = blockIdx.x % BPB;

  const float4* p = (const float4*)img + (size_t)batch * (CH * HW4);

  float sum = 0.0f;
  int nb = 0;  // bright: g >= 0.75 && g <= 1.0
  int nd = 0;  // dark:   g >= 0.0  && g <  0.25
  const float third = 1.0f / 3.0f;

  for (int i = blk * TPB + threadIdx.x; i < HW4; i += BPB * TPB) {
    float4 a = p[i];
    float4 b = p[i + HW4];
    float4 c = p[i + 2 * HW4];
    float g0 = (a.x + b.x + c.x) * third;
    float g1 = (a.y + b.y + c.y) * third;
    float g2 = (a.z + b.z + c.z) * third;
    float g3 = (a.w + b.w + c.w) * third;
    sum += (g0 + g1) + (g2 + g3);
    nb += (g0 >= 0.75f && g0 <= 1.0f) + (g1 >= 0.75f && g1 <= 1.0f) +
          (g2 >= 0.75f && g2 <= 1.0f) + (g3 >= 0.75f && g3 <= 1.0f);
    nd += (g0 >= 0.0f && g0 < 0.25f) + (g1 >= 0.0f && g1 < 0.25f) +
          (g2 >= 0.0f && g2 < 0.25f) + (g3 >= 0.0f && g3 < 0.25f);
  }

  float fb = (float)nb;
  float fd = (float)nd;

  // wave32 butterfly reduce
  for (int off = 16; off > 0; off >>= 1) {
    sum += __shfl_xor(sum, off, 32);
    fb  += __shfl_xor(fb,  off, 32);
    fd  += __shfl_xor(fd,  off, 32);
  }

  __shared__ float ls[TPB / 32], lb[TPB / 32], ld[TPB / 32];
  const int wave = threadIdx.x >> 5;
  const int lane = threadIdx.x & 31;
  if (lane == 0) { ls[wave] = sum; lb[wave] = fb; ld[wave] = fd; }
  __syncthreads();

  if (threadIdx.x == 0) {
    float S = 0.0f, Bc = 0.0f, D = 0.0f;
    for (int w = 0; w < TPB / 32; ++w) { S += ls[w]; Bc += lb[w]; D += ld[w]; }
    const int idx = batch * BPB + blk;
    ws[idx]             = S;   // gray-sum plane
    ws[NPART + idx]     = Bc;  // bright plane
    ws[2 * NPART + idx] = D;   // dark plane
  }
}

// ---------------------------------------------------------------------------
// Kernel 2: one-wave WMMA finalize.
// Reduce the three 16x128 partial matrices over the block (K) dimension with
// V_WMMA_F32_16X16X4_F32:  C += A(16x4) x B(4x16, all ones).
// A layout per ISA (32-bit 16x4): lane L<16 holds row L, {VGPR0=K0, VGPR1=K1};
// lanes 16..31 hold row L-16, {VGPR0=K2, VGPR1=K3}  -> a contiguous float2.
// B = all-ones makes its lane layout immaterial. D column n replicates the
// row sums, so lane 0 spills rows 0..7, lane 16 spills rows 8..15 to LDS.
// Then 16 lanes run the scalar epilogue (dr, avg, gap select, e1/e2).
// ---------------------------------------------------------------------------
__global__ void finalize_kernel(const float* __restrict__ ws,
                                const float* __restrict__ be1,
                                const float* __restrict__ be2,
                                float* __restrict__ out) {
  const int lane = threadIdx.x;      // 0..31, full wave, EXEC all ones
  const int row  = lane & 15;
  const int hi   = lane >> 4;

  const float* sumP = ws;
  const float* brP  = ws + NPART;
  const float* dkP  = ws + 2 * NPART;

  v8f cS = {}, cB = {}, cD = {};
  v2f ones; ones.x = 1.0f; ones.y = 1.0f;

  for (int k0 = 0; k0 < BPB; k0 += 4) {
    const int off = row * BPB + k0 + 2 * hi;
    v2f aS; aS.x = sumP[off]; aS.y = sumP[off + 1];
    v2f aB; aB.x = brP[off];  aB.y = brP[off + 1];
    v2f aD; aD.x = dkP[off];  aD.y = dkP[off + 1];
    cS = __builtin_amdgcn_wmma_f32_16x16x4_f32(false, aS, false, ones,
                                               (short)0, cS, false, false);
    cB = __builtin_amdgcn_wmma_f32_16x16x4_f32(false, aB, false, ones,
                                               (short)0, cB, false, false);
    cD = __builtin_amdgcn_wmma_f32_16x16x4_f32(false, aD, false, ones,
                                               (short)0, cD, false, false);
  }

  __shared__ float sSum[16], sBr[16], sDk[16], sAvg[16], sGap[16];
  if (row == 0) {  // lanes 0 and 16: D VGPR r holds row (hi*8 + r), any column
    for (int r = 0; r < 8; ++r) {
      sSum[hi * 8 + r] = cS[r];
      sBr[hi * 8 + r]  = cB[r];
      sDk[hi * 8 + r]  = cD[r];
    }
  }
  __syncthreads();

  if (lane < 16) {
    const float avg = sSum[lane] * (1.0f / (float)HW);
    const float dr  = sBr[lane] / (sDk[lane] + 1e-5f);
    float gap;
    if (dr > 1.0f && avg > 0.4f && avg < 0.6f)        gap = 1.0f;    // 0.5*2
    else if (avg <= 0.3f)                             gap = 0.25f;   // 0.5*0.5
    else if (avg >= 0.7f)                             gap = 0.25f;
    else if (dr <= 1.0f && avg > 0.3f && avg < 0.7f)  gap = 0.375f;  // 0.5*0.75
    else                                              gap = 0.0f;
    out[lane]      = dr;
    out[16 + lane] = avg;
    out[32 + lane] = gap;
    sAvg[lane] = avg;
    sGap[lane] = gap;
  }
  __syncthreads();

  if (lane < 16) {
    const float bl = sAvg[15];   // last batch element drives e1/e2 (ref quirk)
    const float gl = sGap[15];
    const float b1 = be1[lane];
    const float b2 = be2[lane];
    float e1, e2;
    if (bl <= 0.25f)      { e1 = b1 + 0.5f * gl * 1.7f; e2 = b2 + 0.5f * gl * 1.7f; }
    else if (bl >= 0.75f) { e1 = b1 - 0.5f * gl * 1.7f; e2 = b2 - 0.5f * gl * 1.7f; }
    else                  { e1 = b1 - 0.3f * gl;        e2 = b2 + 0.7f * gl; }
    out[48 + lane] = e1;
    out[64 + lane] = e2;
  }
}

// ---------------------------------------------------------------------------
// Launch: 2048 blocks x 256 threads streaming, then one wave of finalize.
// ws usage: 3 * 2048 floats = 24 KB. Fully deterministic (no atomics; every
// ws slot is overwritten every launch).
// ---------------------------------------------------------------------------
extern "C" void kernel_launch(void* const* d_in, const int* in_sizes, int n_in,
                              void* d_out, int out_size, void* d_ws, size_t ws_size,
                              hipStream_t stream) {
  (void)in_sizes; (void)n_in; (void)out_size; (void)ws_size;
  const float* img = (const float*)d_in[0];
  const float* be1 = (const float*)d_in[1];
  const float* be2 = (const float*)d_in[2];
  float* out = (float*)d_out;
  float* ws  = (float*)d_ws;

  reduce_kernel<<<dim3(BATCH * BPB), dim3(TPB), 0, stream>>>(img, ws);
  finalize_kernel<<<dim3(1), dim3(32), 0, stream>>>(ws, be1, be2, out);
}